// LinearMultihead_64802466562903
// MI455X (gfx1250) — compile-verified
//
#include <hip/hip_runtime.h>

typedef float v2f __attribute__((ext_vector_type(2)));
typedef float v8f __attribute__((ext_vector_type(8)));

#define IN_F   128
#define OUT_F  128
#define NHEADS 64
#define TILE_M 16
#define LDS_STRIDE 132   // 128 + 4 pad -> conflict-free A fragment reads

// ---------------- bucketing kernels ----------------

__global__ void lm_zero_ws(int* __restrict__ ws) {
    int t = blockIdx.x * blockDim.x + threadIdx.x;
    if (t < 128) ws[t] = 0;   // counts[64] + cursors[64]
}

__global__ void lm_count(const int* __restrict__ head_ix, int* __restrict__ counts, int batch) {
    int i = blockIdx.x * blockDim.x + threadIdx.x;
    if (i < batch) atomicAdd(&counts[head_ix[i]], 1);
}

__global__ void lm_prefix(const int* __restrict__ counts, int* __restrict__ offsets) {
    if (threadIdx.x == 0) {
        int acc = 0;
        for (int h = 0; h < NHEADS; ++h) { offsets[h] = acc; acc += counts[h]; }
    }
}

__global__ void lm_scatter(const int* __restrict__ head_ix, const int* __restrict__ offsets,
                           int* __restrict__ cursors, int* __restrict__ rowlist, int batch) {
    int i = blockIdx.x * blockDim.x + threadIdx.x;
    if (i < batch) {
        int h = head_ix[i];
        int pos = atomicAdd(&cursors[h], 1);
        rowlist[offsets[h] + pos] = i;
    }
}

// ---------------- WMMA GEMM kernel ----------------
// grid: (ceil(batch/16), NHEADS), block: 256 (8 wave32s)
// Each block: one 16-row tile of rows sharing head h, full 128 outputs.
// Wave w computes the 16x16 output tile at columns [16w, 16w+16).

__global__ __launch_bounds__(256)
void lm_head_gemm(const float* __restrict__ X, const float* __restrict__ W,
                  const float* __restrict__ Bias,
                  const int* __restrict__ counts, const int* __restrict__ offsets,
                  const int* __restrict__ rowlist, float* __restrict__ out)
{
    __shared__ float sA[TILE_M * LDS_STRIDE];
    __shared__ int   sRows[TILE_M];

    const int h     = blockIdx.y;
    const int cnt   = counts[h];
    const int base0 = blockIdx.x * TILE_M;
    if (base0 >= cnt) return;                    // inactive tile: cheap early exit
    const int nrows = min(TILE_M, cnt - base0);
    const int lbase = offsets[h] + base0;

    const int t = threadIdx.x;

    // Gather 16 rows of x into LDS (each thread: 8 contiguous floats of one row)
    {
        const int rl = min(t >> 4, nrows - 1);       // local row 0..15 (clamped dup for ragged tile)
        const int gr = rowlist[lbase + rl];          // global row index
        if (t < TILE_M) sRows[t] = rowlist[lbase + min(t, nrows - 1)];
        const float4* src = (const float4*)(X + (size_t)gr * IN_F + (t & 15) * 8);
        float4 v0 = src[0];
        float4 v1 = src[1];
        float* dst = &sA[(t >> 4) * LDS_STRIDE + (t & 15) * 8];
        dst[0] = v0.x; dst[1] = v0.y; dst[2] = v0.z; dst[3] = v0.w;
        dst[4] = v1.x; dst[5] = v1.y; dst[6] = v1.z; dst[7] = v1.w;
    }
    __syncthreads();

    const int lane = t & 31;
    const int m    = lane & 15;       // A: row M / B: col N / C: col N
    const int hi   = lane >> 4;       // upper half-wave selector
    const int n0   = (t >> 5) * 16;   // this wave's output column base

    const float* Wh = W + (size_t)h * (IN_F * OUT_F);

    // C initialized with broadcast bias (N = m, independent of row)
    v8f c;
    {
        const float bv = Bias[h * OUT_F + n0 + m];
        #pragma unroll
        for (int r = 0; r < 8; ++r) c[r] = bv;
    }

    // K loop: 32 x V_WMMA_F32_16X16X4_F32
    #pragma unroll
    for (int k0 = 0; k0 < IN_F; k0 += 4) {
        v2f a, b;
        // A fragment: lane -> X_tile[m][k0 + 2*hi], [m][k0 + 2*hi + 1]   (ds_load_b64)
        const float* ap = &sA[m * LDS_STRIDE + k0 + 2 * hi];
        a.x = ap[0];
        a.y = ap[1];
        // B fragment: lane -> W[h][k0 + 2*hi][n0+m], W[h][k0 + 2*hi + 1][n0+m]
        const float* bp = Wh + (size_t)(k0 + 2 * hi) * OUT_F + n0 + m;
        b.x = bp[0];
        b.y = bp[OUT_F];
        c = __builtin_amdgcn_wmma_f32_16x16x4_f32(
                /*neg_a=*/false, a, /*neg_b=*/false, b,
                /*c_mod=*/(short)0, c, /*reuse_a=*/false, /*reuse_b=*/false);
    }

    // Store D: VGPR r holds (M = r + 8*hi, N = m); mask ragged rows
    #pragma unroll
    for (int r = 0; r < 8; ++r) {
        const int row_local = r + 8 * hi;
        if (row_local < nrows) {
            out[(size_t)sRows[row_local] * OUT_F + n0 + m] = c[r];
        }
    }
}

// ---------------- launch ----------------

extern "C" void kernel_launch(void* const* d_in, const int* in_sizes, int n_in,
                              void* d_out, int out_size, void* d_ws, size_t ws_size,
                              hipStream_t stream) {
    const float* X       = (const float*)d_in[0];   // [B, 128]
    const float* W       = (const float*)d_in[1];   // [64, 128, 128]
    const float* Bias    = (const float*)d_in[2];   // [64, 128]
    const int*   head_ix = (const int*)d_in[3];     // [B]
    float*       out     = (float*)d_out;

    const int batch = in_sizes[0] / IN_F;           // 8192

    int* ws      = (int*)d_ws;
    int* counts  = ws;          // [64]
    int* cursors = ws + 64;     // [64]
    int* offsets = ws + 128;    // [64]
    int* rowlist = ws + 192;    // [batch]

    lm_zero_ws<<<1, 128, 0, stream>>>(ws);
    lm_count<<<(batch + 255) / 256, 256, 0, stream>>>(head_ix, counts, batch);
    lm_prefix<<<1, 64, 0, stream>>>(counts, offsets);
    lm_scatter<<<(batch + 255) / 256, 256, 0, stream>>>(head_ix, offsets, cursors, rowlist, batch);

    dim3 grid((batch + TILE_M - 1) / TILE_M, NHEADS);
    lm_head_gemm<<<grid, 256, 0, stream>>>(X, W, Bias, counts, offsets, rowlist, out);
}